// mixprop_gat_init_36292473651934
// MI455X (gfx1250) — compile-verified
//
#include <hip/hip_runtime.h>
#include <hip/hip_bf16.h>
#include <math.h>

#define CCH 64
#define NEG_SLOPE 0.2f
#define MIX_ALPHA 0.05f
#define EPSV 1e-16f

typedef float v2f __attribute__((ext_vector_type(2)));
typedef float v8f __attribute__((ext_vector_type(8)));
typedef unsigned int v4u __attribute__((ext_vector_type(4)));
typedef int v4i __attribute__((ext_vector_type(4)));
typedef int v8i __attribute__((ext_vector_type(8)));

// ---------------------------------------------------------------------------
// CDNA5 fp32 WMMA: D(16x16) = A(16x4) * B(4x16) + C   (wave32, striped layout)
// A: lane holds M = lane&15, K = 2*(lane>>4) + {v0,v1}
// B: lane holds N = lane&15, K = 2*(lane>>4) + {v0,v1}
// C/D: vgpr j -> row (j + 8*(lane>>4)), col = lane&15
// ---------------------------------------------------------------------------
__device__ __forceinline__ v8f wmma_f32_16x16x4(v2f a, v2f b, v8f c) {
#if __has_builtin(__builtin_amdgcn_wmma_f32_16x16x4_f32)
  return __builtin_amdgcn_wmma_f32_16x16x4_f32(false, a, false, b, (short)0, c,
                                               false, false);
#else
  (void)a; (void)b;
  return c;
#endif
}

// ---------------------------------------------------------------------------
// Tensor Data Mover: DMA a 64x64 fp32 tile (16 KB, contiguous, stride 64)
// from global memory into LDS.  D# per CDNA5 ISA:
//  group0: count=1 | lds_addr | global_addr[56:0] | type=2 ("image")
//  group1: data_size=2 (4B) | tensor_dim0/1=64 | tile_dim0/1=64 | stride0=64
//  groups 2/3: zero (2-D tensor)
// Caller: one wave issues, then s_wait_tensorcnt(0), then workgroup barrier.
// ---------------------------------------------------------------------------
#if __has_builtin(__builtin_amdgcn_tensor_load_to_lds)
#define HAVE_TDM 1
#endif

__device__ __forceinline__ void tdm_load_tile_64x64_f32(const float* gsrc,
                                                        unsigned lds_off) {
#ifdef HAVE_TDM
  const unsigned long long ga = (unsigned long long)(uintptr_t)gsrc;
  v4u g0;
  g0.x = 1u;                                   // count=1 (valid user D#)
  g0.y = lds_off;                              // LDS byte address
  g0.z = (unsigned)(ga & 0xFFFFFFFFu);         // global_addr[31:0]
  g0.w = (unsigned)((ga >> 32) & 0x01FFFFFFu)  // global_addr[56:32]
         | (2u << 30);                         // type = 2
  v8i g1 = {};
  g1[0] = 0x00020000;          // workgroup_mask=0, data_size=2 (4 bytes)
  g1[1] = (int)(64u << 16);    // tensor_dim0[15:0] = 64  (bits 63:48)
  g1[2] = (int)(64u << 16);    // tensor_dim1[15:0] = 64  (bits 95:80)
  g1[3] = (int)(64u << 16);    // tile_dim0 = 64          (bits 127:112)
  g1[4] = 64;                  // tile_dim1 = 64          (bits 143:128)
  g1[5] = 64;                  // tensor_dim0_stride = 64 (bits 191:160)
  v4i gz = {};
#if __clang_major__ >= 23
  v8i gz8 = {};
  __builtin_amdgcn_tensor_load_to_lds(g0, g1, gz, gz, gz8, 0);
#else
  __builtin_amdgcn_tensor_load_to_lds(g0, g1, gz, gz, 0);
#endif
  __builtin_amdgcn_s_wait_tensorcnt(0);
#else
  (void)gsrc; (void)lds_off;
#endif
}

// xw[N x 64] = H[N x 64] @ W[64 x 64]; W staged in LDS via TDM
__global__ void __launch_bounds__(128)
gemm64_wmma_kernel(const float* __restrict__ A, const float* __restrict__ B,
                   float* __restrict__ out, int N) {
  __shared__ float Bs[CCH * CCH];
  const int tid = threadIdx.x;
#ifdef HAVE_TDM
  if (tid < 32) {
    tdm_load_tile_64x64_f32(B, (unsigned)(uintptr_t)(void*)&Bs[0]);
  }
#else
  {
    const float4* sp = (const float4*)B;
    float4* dp = (float4*)Bs;
    for (int i = tid; i < CCH * CCH / 4; i += 128) dp[i] = sp[i];
  }
#endif
  __syncthreads();

  const int wave = tid >> 5;
  const int lane = tid & 31;
  const int row0 = (blockIdx.x * 4 + wave) * 16;
  if (row0 >= N) return;  // only barrier already passed
  const int half = lane >> 4;
  const int ml = lane & 15;

  int arowi = row0 + ml;
  if (arowi >= N) arowi = N - 1;
  const float* arow = A + (size_t)arowi * CCH;

  v8f acc[4] = {};
#pragma unroll
  for (int k0 = 0; k0 < CCH; k0 += 4) {
    const int kb = k0 + 2 * half;
    v2f a;
    a.x = arow[kb];
    a.y = arow[kb + 1];
#pragma unroll
    for (int nt = 0; nt < 4; ++nt) {
      v2f b;
      const int nc = nt * 16 + ml;
      b.x = Bs[kb * CCH + nc];
      b.y = Bs[(kb + 1) * CCH + nc];
      acc[nt] = wmma_f32_16x16x4(a, b, acc[nt]);
    }
  }

  const int rbase = row0 + 8 * half;
#pragma unroll
  for (int j = 0; j < 8; ++j) {
    const int r = rbase + j;
    if (r < N) {
#pragma unroll
      for (int nt = 0; nt < 4; ++nt)
        out[(size_t)r * CCH + nt * 16 + ml] = acc[nt][j];
    }
  }
}

// e_src[i] = xw[i,:] . a_src ; e_dst[i] = xw[i,:] . a_dst
__global__ void __launch_bounds__(256)
attn_logits_kernel(const float* __restrict__ xw, const float* __restrict__ asrc,
                   const float* __restrict__ adst, float* __restrict__ es,
                   float* __restrict__ ed, int N) {
  const int i = blockIdx.x * blockDim.x + threadIdx.x;
  if (i >= N) return;
  const float4* xr = (const float4*)(xw + (size_t)i * CCH);
  const float4* a4 = (const float4*)asrc;
  const float4* b4 = (const float4*)adst;
  float s1 = 0.f, s2 = 0.f;
#pragma unroll
  for (int j = 0; j < CCH / 4; ++j) {
    const float4 v = xr[j];
    const float4 a = a4[j];
    const float4 b = b4[j];
    s1 += v.x * a.x + v.y * a.y + v.z * a.z + v.w * a.w;
    s2 += v.x * b.x + v.y * b.y + v.z * b.z + v.w * b.w;
  }
  es[i] = s1;
  ed[i] = s2;
}

__global__ void __launch_bounds__(256)
layer_init_kernel(float* __restrict__ m, float* __restrict__ s,
                  float* __restrict__ agg, int N) {
  const int i = blockIdx.x * blockDim.x + threadIdx.x;
  if (i < N * CCH) agg[i] = 0.f;
  if (i < N) {
    m[i] = -INFINITY;
    s[i] = 0.f;
  }
}

// monotone int/uint trick for IEEE float atomic max (works with -inf init)
__device__ __forceinline__ void atomicMaxF(float* addr, float val) {
  if (val >= 0.f)
    atomicMax((int*)addr, __float_as_int(val));
  else
    atomicMin((unsigned int*)addr, __float_as_uint(val));
}

__global__ void __launch_bounds__(256)
edge_max_kernel(const long long* __restrict__ ei, const float* __restrict__ es,
                const float* __restrict__ ed, float* __restrict__ m,
                float* __restrict__ ee, int E, int ET) {
  const int i = blockIdx.x * blockDim.x + threadIdx.x;
  if (i >= ET) return;
  long long sn, dn;
  if (i < E) {
    sn = ei[i];
    dn = ei[E + i];
  } else {
    sn = dn = (long long)(i - E);  // self-loop
  }
  float e = es[sn] + ed[dn];
  e = (e > 0.f) ? e : NEG_SLOPE * e;  // leaky_relu
  ee[i] = e;
  atomicMaxF(&m[dn], e);
}

__global__ void __launch_bounds__(256)
edge_expsum_kernel(const long long* __restrict__ ei, const float* __restrict__ ee,
                   const float* __restrict__ m, float* __restrict__ s,
                   float* __restrict__ ex, int E, int ET) {
  const int i = blockIdx.x * blockDim.x + threadIdx.x;
  if (i >= ET) return;
  const long long dn = (i < E) ? ei[E + i] : (long long)(i - E);
  const float v = __expf(ee[i] - m[dn]);
  ex[i] = v;
  atomicAdd(&s[dn], v);
}

// 64 lanes per edge: lane c scatters attn * xw[src, c] into agg[dst, c]
__global__ void __launch_bounds__(256)
edge_agg_kernel(const long long* __restrict__ ei, const float* __restrict__ ex,
                const float* __restrict__ s, const float* __restrict__ xw,
                float* __restrict__ agg, int E, int ET) {
  const int e = blockIdx.x * 4 + (threadIdx.x >> 6);
  const int c = threadIdx.x & 63;
  if (e >= ET) return;
  long long sn, dn;
  if (e < E) {
    sn = ei[e];
    dn = ei[E + e];
  } else {
    sn = dn = (long long)(e - E);
  }
  const float attn = ex[e] / (s[dn] + EPSV);
  atomicAdd(&agg[(size_t)dn * CCH + c], attn * xw[(size_t)sn * CCH + c]);
}

__global__ void __launch_bounds__(256)
mix_kernel(const float* __restrict__ x, const float* __restrict__ agg,
           const float* __restrict__ bias, float* __restrict__ h, int N) {
  const int i = blockIdx.x * blockDim.x + threadIdx.x;
  if (i >= N * CCH) return;
  const int c = i & (CCH - 1);
  h[i] = MIX_ALPHA * x[i] + (1.f - MIX_ALPHA) * (agg[i] + bias[c]);
}

// out[N x 64] = concat(h0,h1,h2,h3)[N x 256] @ mlp_w[256 x 64] + mlp_b
__global__ void __launch_bounds__(128)
final_gemm_kernel(const float* __restrict__ h0, const float* __restrict__ h1,
                  const float* __restrict__ h2, const float* __restrict__ h3,
                  const float* __restrict__ Wm, const float* __restrict__ bias,
                  float* __restrict__ out, int N) {
  __shared__ float Bs[CCH * CCH];
  const int tid = threadIdx.x;
  const int lane = tid & 31;
  const int row0 = (blockIdx.x * 4 + (tid >> 5)) * 16;
  const int half = lane >> 4;
  const int ml = lane & 15;

  int arowi = row0 + ml;
  if (arowi >= N) arowi = N - 1;
  const float* srcs[4] = {h0, h1, h2, h3};

  v8f acc[4] = {};
  for (int kb = 0; kb < 4; ++kb) {
    __syncthreads();
#ifdef HAVE_TDM
    if (tid < 32) {
      tdm_load_tile_64x64_f32(Wm + (size_t)kb * CCH * CCH,
                              (unsigned)(uintptr_t)(void*)&Bs[0]);
    }
#else
    {
      const float4* sp = (const float4*)(Wm + (size_t)kb * CCH * CCH);
      float4* dp = (float4*)Bs;
      for (int i2 = tid; i2 < CCH * CCH / 4; i2 += 128) dp[i2] = sp[i2];
    }
#endif
    __syncthreads();
    if (kb < 3) __builtin_prefetch(Wm + (size_t)(kb + 1) * CCH * CCH, 0, 0);

    const float* arow = srcs[kb] + (size_t)arowi * CCH;
#pragma unroll
    for (int k0 = 0; k0 < CCH; k0 += 4) {
      const int kbase = k0 + 2 * half;
      v2f a;
      a.x = arow[kbase];
      a.y = arow[kbase + 1];
#pragma unroll
      for (int nt = 0; nt < 4; ++nt) {
        v2f b;
        const int nc = nt * 16 + ml;
        b.x = Bs[kbase * CCH + nc];
        b.y = Bs[(kbase + 1) * CCH + nc];
        acc[nt] = wmma_f32_16x16x4(a, b, acc[nt]);
      }
    }
  }

  const int rbase = row0 + 8 * half;
#pragma unroll
  for (int j = 0; j < 8; ++j) {
    const int r = rbase + j;
    if (r < N) {
#pragma unroll
      for (int nt = 0; nt < 4; ++nt)
        out[(size_t)r * CCH + nt * 16 + ml] = acc[nt][j] + bias[nt * 16 + ml];
    }
  }
}

extern "C" void kernel_launch(void* const* d_in, const int* in_sizes, int n_in,
                              void* d_out, int out_size, void* d_ws, size_t ws_size,
                              hipStream_t stream) {
  const float* x = (const float*)d_in[0];
  const long long* ei = (const long long*)d_in[1];  // int64 edge_index [2, E]
  const float* W = (const float*)d_in[2];
  const float* asrc = (const float*)d_in[3];
  const float* adst = (const float*)d_in[4];
  const float* bias = (const float*)d_in[5];
  const float* mlpw = (const float*)d_in[6];
  const float* mlpb = (const float*)d_in[7];
  float* out = (float*)d_out;

  const int N = in_sizes[0] / CCH;
  const int E = in_sizes[1] / 2;
  const int G = in_sizes[2] / (CCH * CCH);  // 3
  const int ET = E + N;                     // edges + self-loops

  // workspace layout (floats)
  float* ws = (float*)d_ws;
  size_t o = 0;
  float* xw = ws + o;  o += (size_t)N * CCH;
  float* agg = ws + o; o += (size_t)N * CCH;
  float* es = ws + o;  o += (size_t)N;
  float* ed = ws + o;  o += (size_t)N;
  float* mv = ws + o;  o += (size_t)N;
  float* sv = ws + o;  o += (size_t)N;
  float* ee = ws + o;  o += (size_t)ET;
  float* ex = ws + o;  o += (size_t)ET;
  float* hs = ws + o;  // G * N * CCH

  const dim3 blkG(128);
  const dim3 grdG((N + 63) / 64);
  const int t256 = 256;
  const int grdN = (N + 255) / 256;
  const int grdNC = (N * CCH + 255) / 256;
  const int grdE = (ET + 255) / 256;
  const int grdEA = (ET + 3) / 4;

  const float* hcur = x;
  for (int i = 0; i < G; ++i) {
    float* hnext = hs + (size_t)i * N * CCH;
    gemm64_wmma_kernel<<<grdG, blkG, 0, stream>>>(hcur, W + (size_t)i * CCH * CCH, xw, N);
    attn_logits_kernel<<<grdN, t256, 0, stream>>>(xw, asrc + i * CCH, adst + i * CCH,
                                                  es, ed, N);
    layer_init_kernel<<<grdNC, t256, 0, stream>>>(mv, sv, agg, N);
    edge_max_kernel<<<grdE, t256, 0, stream>>>(ei, es, ed, mv, ee, E, ET);
    edge_expsum_kernel<<<grdE, t256, 0, stream>>>(ei, ee, mv, sv, ex, E, ET);
    edge_agg_kernel<<<grdEA, t256, 0, stream>>>(ei, ex, sv, xw, agg, E, ET);
    mix_kernel<<<grdNC, t256, 0, stream>>>(x, agg, bias + i * CCH, hnext, N);
    hcur = hnext;
  }

  const float* h1 = hs;
  const float* h2 = hs + (size_t)N * CCH;
  const float* h3 = hs + 2 * (size_t)N * CCH;
  final_gemm_kernel<<<grdG, blkG, 0, stream>>>(x, h1, h2, h3, mlpw, mlpb, out, N);
}